// FastGTNLayer_85220741087423
// MI455X (gfx1250) — compile-verified
//
#include <hip/hip_runtime.h>
#include <math.h>

typedef __attribute__((ext_vector_type(2))) float v2f;
typedef __attribute__((ext_vector_type(8))) float v8f;

#define N_NODES 8192
#define N_EDGES 4
#define IN_C    64
#define OUT_C   64
#define KSPLIT  16            // split-K factor for occupancy (4096 waves total)
#define ROWS_PER_WAVE 32      // 2 WMMA M-tiles per wave

// Per-thread (uniform) computation of scale[e] = sum_c softmax(weight, axis=0)[e,c].
// weight is [N_EDGES, 2]; 8 scalar loads, negligible.
__device__ __forceinline__ void edge_scales(const float* __restrict__ W, float s[N_EDGES]) {
    float w[N_EDGES][2];
#pragma unroll
    for (int e = 0; e < N_EDGES; ++e) {
        w[e][0] = W[e * 2 + 0];
        w[e][1] = W[e * 2 + 1];
    }
#pragma unroll
    for (int e = 0; e < N_EDGES; ++e) s[e] = 0.0f;
#pragma unroll
    for (int c = 0; c < 2; ++c) {
        float m = w[0][c];
#pragma unroll
        for (int e = 1; e < N_EDGES; ++e) m = fmaxf(m, w[e][c]);
        float ex[N_EDGES];
        float sum = 0.0f;
#pragma unroll
        for (int e = 0; e < N_EDGES; ++e) { ex[e] = __expf(w[e][c] - m); sum += ex[e]; }
        float inv = 1.0f / sum;
#pragma unroll
        for (int e = 0; e < N_EDGES; ++e) s[e] += ex[e] * inv;
    }
}

// Fused: H_partial[32x64 tile] += (sum_e scale[e]*A[e]) @ X over a K-chunk.
// One wave per block. f32 WMMA 16x16x4.
// A-fragment layout (32-bit A 16x4): lane&15 = M row, VGPR0/1 = K+0/K+1,
// lanes 16-31 hold K+2/K+3. B/C/D: rows striped across lanes within a VGPR.
// A is a 1 GiB single-use stream -> non-temporal loads so the L2 keeps the
// heavily reused X (2 MiB, re-read by every wave) and the H accumulator hot.
__global__ __launch_bounds__(32)
void gtn_agg_gemm(const float* __restrict__ A, const float* __restrict__ X,
                  const float* __restrict__ W, float* __restrict__ Hp) {
    const int lane = threadIdx.x;
    const int half = lane >> 4;        // 0: K+0/1, 1: K+2/3
    const int ln   = lane & 15;
    const int rowBase = blockIdx.x * ROWS_PER_WAVE;
    const int kChunk  = N_NODES / KSPLIT;
    const int kBegin  = blockIdx.y * kChunk;
    const int kEnd    = kBegin + kChunk;

    float s[N_EDGES];
    edge_scales(W, s);

    v8f acc[2][4];
#pragma unroll
    for (int mt = 0; mt < 2; ++mt)
#pragma unroll
        for (int ct = 0; ct < 4; ++ct)
            acc[mt][ct] = (v8f)(0.0f);

    const long long edgeStride = (long long)N_NODES * (long long)N_NODES;
    const float* __restrict__ arow0 = A + (long long)(rowBase + ln) * N_NODES;
    const float* __restrict__ arow1 = A + (long long)(rowBase + 16 + ln) * N_NODES;

    for (int k = kBegin; k < kEnd; k += 4) {
        const int kk = k + 2 * half;

        // B fragments: X[kk..kk+1, ct*16+ln] (L2-resident; default RT policy)
        v2f b[4];
#pragma unroll
        for (int ct = 0; ct < 4; ++ct) {
            b[ct].x = X[(long long)kk * IN_C + ct * 16 + ln];
            b[ct].y = X[(long long)(kk + 1) * IN_C + ct * 16 + ln];
        }

#pragma unroll
        for (int mt = 0; mt < 2; ++mt) {
            const float* __restrict__ ap = (mt ? arow1 : arow0) + kk;
            // aggregate the 4 edge matrices on the fly (A read exactly once,
            // streamed with TH_NT so it never displaces the reusable set)
            v2f a0 = __builtin_nontemporal_load((const v2f*)(ap));
            v2f a1 = __builtin_nontemporal_load((const v2f*)(ap + edgeStride));
            v2f a2 = __builtin_nontemporal_load((const v2f*)(ap + 2 * edgeStride));
            v2f a3 = __builtin_nontemporal_load((const v2f*)(ap + 3 * edgeStride));
            v2f a  = a0 * s[0] + a1 * s[1] + a2 * s[2] + a3 * s[3];

#pragma unroll
            for (int ct = 0; ct < 4; ++ct) {
                acc[mt][ct] = __builtin_amdgcn_wmma_f32_16x16x4_f32(
                    false, a, false, b[ct], (short)0, acc[mt][ct], false, false);
            }
        }
    }

    // Accumulate split-K partials: C/D layout -> row = v + 8*half, col = ln
#pragma unroll
    for (int mt = 0; mt < 2; ++mt)
#pragma unroll
        for (int ct = 0; ct < 4; ++ct)
#pragma unroll
            for (int v = 0; v < 8; ++v) {
                int row = rowBase + mt * 16 + v + 8 * half;
                int col = ct * 16 + ln;
                atomicAdd(&Hp[(long long)row * IN_C + col], acc[mt][ct][v]);
            }
}

// out = H @ lin_w^T + lin_b + bias, via f32 WMMA. B[k=c][n=o] = lin_w[o*64+c]
// -> contiguous float2 fragments.
__global__ __launch_bounds__(32)
void gtn_linear(const float* __restrict__ H, const float* __restrict__ Wl,
                const float* __restrict__ bl, const float* __restrict__ bias,
                float* __restrict__ out) {
    const int lane = threadIdx.x;
    const int half = lane >> 4;
    const int ln   = lane & 15;
    const int rowBase = blockIdx.x * ROWS_PER_WAVE;

    v8f acc[2][4];
#pragma unroll
    for (int mt = 0; mt < 2; ++mt)
#pragma unroll
        for (int ct = 0; ct < 4; ++ct)
            acc[mt][ct] = (v8f)(0.0f);

#pragma unroll
    for (int k = 0; k < IN_C; k += 4) {
        const int kk = k + 2 * half;
        v2f b[4];
#pragma unroll
        for (int ct = 0; ct < 4; ++ct)
            b[ct] = *(const v2f*)(Wl + (ct * 16 + ln) * IN_C + kk);

#pragma unroll
        for (int mt = 0; mt < 2; ++mt) {
            v2f a = *(const v2f*)(H + (long long)(rowBase + mt * 16 + ln) * IN_C + kk);
#pragma unroll
            for (int ct = 0; ct < 4; ++ct) {
                acc[mt][ct] = __builtin_amdgcn_wmma_f32_16x16x4_f32(
                    false, a, false, b[ct], (short)0, acc[mt][ct], false, false);
            }
        }
    }

#pragma unroll
    for (int mt = 0; mt < 2; ++mt)
#pragma unroll
        for (int ct = 0; ct < 4; ++ct)
#pragma unroll
            for (int v = 0; v < 8; ++v) {
                int row = rowBase + mt * 16 + v + 8 * half;
                int col = ct * 16 + ln;
                out[(long long)row * OUT_C + col] = acc[mt][ct][v] + bl[col] + bias[col];
            }
}

extern "C" void kernel_launch(void* const* d_in, const int* in_sizes, int n_in,
                              void* d_out, int out_size, void* d_ws, size_t ws_size,
                              hipStream_t stream) {
    const float* A    = (const float*)d_in[0];  // [4, 8192, 8192]
    const float* X    = (const float*)d_in[1];  // [8192, 64]
    const float* W    = (const float*)d_in[2];  // [4, 2]
    const float* lw   = (const float*)d_in[3];  // [64, 64]
    const float* lb   = (const float*)d_in[4];  // [64]
    const float* bias = (const float*)d_in[5];  // [64]
    float* out = (float*)d_out;                 // [8192, 64] fp32
    float* Hws = (float*)d_ws;                  // 2 MiB partial-H accumulator

    const size_t hbytes = (size_t)N_NODES * IN_C * sizeof(float);
    hipMemsetAsync(d_ws, 0, hbytes, stream);    // zero split-K accumulator (capture-safe)

    dim3 grid(N_NODES / ROWS_PER_WAVE, KSPLIT);
    gtn_agg_gemm<<<grid, 32, 0, stream>>>(A, X, W, Hws);

    gtn_linear<<<N_NODES / ROWS_PER_WAVE, 32, 0, stream>>>(Hws, lw, lb, bias, out);
}